// DynamicSoftLabelAssigner_mmdet2_74406013436315
// MI455X (gfx1250) — compile-verified
//
#include <hip/hip_runtime.h>
#include <hip/hip_bf16.h>
#include <math.h>

typedef float v2f __attribute__((ext_vector_type(2)));
typedef float v8f __attribute__((ext_vector_type(8)));
typedef unsigned int u32x4 __attribute__((ext_vector_type(4)));
typedef unsigned int u32x8 __attribute__((ext_vector_type(8)));

#define KSEL 13
#define INF_COST 1e8f
#define CDIM 80
#define KSTEPS (CDIM / 4)

// ============================================================================
// Kernel 1a (specialized C==80): logits[n,g] = pred_scores[n, labels[g]]
// as S(Nx80) @ onehot(80xG) via V_WMMA_F32_16X16X4_F32.
// One wave owns a 16-row mTile; all 20 A fragments (40 VGPRs) stay resident
// in registers while the wave sweeps every 16-col gTile -> 16x less A traffic,
// 320 register-fed WMMAs per wave. Output g-major: logits_t[g*n + m].
// ============================================================================
__global__ void logits_wmma80_kernel(const float* __restrict__ scores,
                                     const int* __restrict__ labels,
                                     float* __restrict__ logits_t,
                                     int n, int g) {
    const int wave = blockIdx.x * (blockDim.x >> 5) + (threadIdx.x >> 5);
    const int lane = threadIdx.x & 31;
    const int m0 = wave << 4;
    if (m0 >= n) return;
    const int l16 = lane & 15;
    const int hi  = lane >> 4;

    const float* srow = scores + (size_t)(m0 + l16) * CDIM;
    v2f a[KSTEPS];
#pragma unroll
    for (int k = 0; k < KSTEPS; ++k) {
        const int ka = 4 * k + 2 * hi;       // A layout: lanes0-15 K=0..1, lanes16-31 K=2..3
        a[k].x = srow[ka];
        a[k].y = srow[ka + 1];
    }

    for (int g0 = 0; g0 < g; g0 += 16) {
        const int lab = labels[g0 + l16];
        v8f acc = {};
#pragma unroll
        for (int k = 0; k < KSTEPS; ++k) {
            const int ka = 4 * k + 2 * hi;
            v2f b;                            // one-hot column fragment
            b.x = (lab == ka)     ? 1.0f : 0.0f;
            b.y = (lab == ka + 1) ? 1.0f : 0.0f;
            acc = __builtin_amdgcn_wmma_f32_16x16x4_f32(
                false, a[k], false, b, (short)0, acc, false, false);
        }
#pragma unroll
        for (int r = 0; r < 8; ++r) {         // D: VGPR r -> row m0+r+8*hi, col g0+l16
            logits_t[(size_t)(g0 + l16) * (size_t)n + (m0 + r + 8 * hi)] = acc[r];
        }
    }
}

// ============================================================================
// Kernel 1b (generic C): same computation, A reloaded per tile.
// ============================================================================
__global__ void logits_wmma_kernel(const float* __restrict__ scores,
                                   const int* __restrict__ labels,
                                   float* __restrict__ logits_t,
                                   int n, int g, int c) {
    const int wave = blockIdx.x * (blockDim.x >> 5) + (threadIdx.x >> 5);
    const int lane = threadIdx.x & 31;
    const int gTiles = g >> 4;
    const int mTile = wave / gTiles;
    const int gTile = wave % gTiles;
    const int m0 = mTile << 4;
    if (m0 >= n) return;
    const int g0 = gTile << 4;
    const int l16 = lane & 15;
    const int hi  = lane >> 4;

    const int lab = labels[g0 + l16];
    const float* srow = scores + (size_t)(m0 + l16) * (size_t)c;

    v8f acc = {};
    for (int c0 = 0; c0 + 3 < c; c0 += 4) {
        const int ka = c0 + 2 * hi;
        v2f a;
        a.x = srow[ka];
        a.y = srow[ka + 1];
        v2f b;
        b.x = (lab == ka)     ? 1.0f : 0.0f;
        b.y = (lab == ka + 1) ? 1.0f : 0.0f;
        acc = __builtin_amdgcn_wmma_f32_16x16x4_f32(
            false, a, false, b, (short)0, acc, false, false);
    }
#pragma unroll
    for (int r = 0; r < 8; ++r) {
        logits_t[(size_t)(g0 + l16) * (size_t)n + (m0 + r + 8 * hi)] = acc[r];
    }
}

// ============================================================================
// Kernel 2: per-prior IoU / validity / cost. GT boxes staged into LDS via the
// Tensor Data Mover (TENSOR_LOAD_TO_LDS, 1-D descriptor), drained with
// s_wait_tensorcnt. ious_t / cost_t stored g-major.
// ============================================================================
__global__ void cost_kernel(const float* __restrict__ priors,
                            const float* __restrict__ pred_bboxes,
                            const float* __restrict__ gt_bboxes,
                            const float* __restrict__ logits_t,
                            float* __restrict__ ious_t,
                            float* __restrict__ cost_t,
                            int* __restrict__ valid_mask,
                            int n, int g) {
    __shared__ float4 s_gt[256];
    const int tid = threadIdx.x;

    // --- TDM: DMA gt_bboxes (g*4 floats) into LDS; wave 0 issues, all wait ---
    if (tid < 32) {
        const unsigned elems = (unsigned)(g * 4);              // f32 elements
        const unsigned lds   = (unsigned)(size_t)(&s_gt[0]);   // LDS byte offset
        const unsigned long long ga = (unsigned long long)(size_t)gt_bboxes;

        u32x4 d0;                                  // D# group 0
        d0.x = 1u;                                 // count=1, user mode
        d0.y = lds;                                // lds_addr [63:32]
        d0.z = (unsigned)(ga & 0xffffffffull);     // global_addr [95:64]
        d0.w = (unsigned)((ga >> 32) & 0x1ffffffull) | (2u << 30); // addr[56:32] | type=2

        u32x8 d1;                                  // D# group 1
        d1.s0 = (2u << 16);                        // workgroup_mask=0, data_size=4B
        d1.s1 = (elems & 0xffffu) << 16;           // tensor_dim0[15:0]
        d1.s2 = ((elems >> 16) & 0xffffu) | (1u << 16); // tensor_dim0[31:16] | tensor_dim1=1
        d1.s3 = (elems & 0xffffu) << 16;           // tensor_dim1[31:16]=0 | tile_dim0
        d1.s4 = 1u;                                // tile_dim1=1, tile_dim2=0
        d1.s5 = elems;                             // tensor_dim0_stride[31:0]
        d1.s6 = 0u;                                // stride[47:32]=0, dim1_stride lo=0
        d1.s7 = 0u;                                // dim1_stride hi=0
        asm volatile("tensor_load_to_lds %0, %1" :: "s"(d0), "s"(d1) : "memory");
        __builtin_amdgcn_s_wait_tensorcnt(0);
    }
    __syncthreads();

    const int i = blockIdx.x * blockDim.x + tid;
    if (i >= n) return;

    const float px = priors[i * 4 + 0];
    const float py = priors[i * 4 + 1];
    const float stride = priors[i * 4 + 2];
    const float4 pb = ((const float4*)pred_bboxes)[i];
    const float area_p = (pb.z - pb.x) * (pb.w - pb.y);

    int valid = 0;
    // Pass 1: IoU + in-gts validity
    for (int j = 0; j < g; ++j) {
        const float4 gb = s_gt[j];
        const float mind = fminf(fminf(px - gb.x, py - gb.y),
                                 fminf(gb.z - px, gb.w - py));
        const bool pad = (gb.x + gb.y + gb.z + gb.w) > 0.0f;
        valid |= (mind > 0.0f) && pad;

        const float area_g = (gb.z - gb.x) * (gb.w - gb.y);
        float iw = fminf(pb.z, gb.z) - fmaxf(pb.x, gb.x);
        float ih = fminf(pb.w, gb.w) - fmaxf(pb.y, gb.y);
        iw = fmaxf(iw, 0.0f);
        ih = fmaxf(ih, 0.0f);
        const float inter = iw * ih;
        const float uni = fmaxf(area_p + area_g - inter, 1e-6f);
        ious_t[(size_t)j * (size_t)n + i] = inter / uni;
    }
    valid_mask[i] = valid;

    // Pass 2: cost
    for (int j = 0; j < g; ++j) {
        const float4 gb = s_gt[j];
        const float gcx = (gb.x + gb.z) * 0.5f;
        const float gcy = (gb.y + gb.w) * 0.5f;
        const float dx = px - gcx, dy = py - gcy;
        float d = sqrtf(dx * dx + dy * dy) / stride;
        d = valid ? d : 0.0f;
        const float soft = powf(10.0f, d - 3.0f);

        const float iou = ious_t[(size_t)j * (size_t)n + i];
        const float iou_cost = -logf(iou + 1e-7f) * 3.0f;

        const float logit = logits_t[(size_t)j * (size_t)n + i];
        const float sig = 1.0f / (1.0f + expf(-logit));
        const float dlt = iou - sig;
        const float scale = dlt * dlt;
        const float bce = fmaxf(logit, 0.0f) - logit * iou +
                          log1pf(expf(-fabsf(logit)));
        const float cv = bce * scale + iou_cost + soft;
        cost_t[(size_t)j * (size_t)n + i] = valid ? cv : INF_COST;
    }
}

// ============================================================================
// Kernel 3: per-GT top-13 (IoU values for dynamic-k; smallest-cost indices).
// One 256-thread block per GT; register sorted lists + LDS tree merge.
// ============================================================================
__global__ void topk_kernel(const float* __restrict__ ious_t,
                            const float* __restrict__ cost_t,
                            int* __restrict__ dyn_ks,
                            int* __restrict__ cand_idx,
                            int n) {
    const int gcol = blockIdx.x;
    const int tid = threadIdx.x;
    const int T = blockDim.x;   // 256

    float liou[KSEL];
    float lc[KSEL];
    int   li[KSEL];
#pragma unroll
    for (int k = 0; k < KSEL; ++k) { liou[k] = -3e38f; lc[k] = 3e38f; li[k] = -1; }

    const float* irow = ious_t + (size_t)gcol * (size_t)n;
    const float* crow = cost_t + (size_t)gcol * (size_t)n;

    for (int i = tid; i < n; i += T) {
        __builtin_prefetch(&irow[i + 4 * T], 0, 0);   // global_prefetch_b8
        __builtin_prefetch(&crow[i + 4 * T], 0, 0);
        const float v = irow[i];
        if (v > liou[KSEL - 1]) {           // bubble-insert, descending
            float cur = v;
#pragma unroll
            for (int q = 0; q < KSEL; ++q) {
                const float mx = fmaxf(liou[q], cur);
                const float mn = fminf(liou[q], cur);
                liou[q] = mx;
                cur = mn;
            }
        }
        const float cv = crow[i];
        if (cv < lc[KSEL - 1]) {            // bubble-insert, ascending, with index
            float cc = cv; int ci = i;
#pragma unroll
            for (int q = 0; q < KSEL; ++q) {
                const bool sw = cc < lc[q];
                const float tf = lc[q];
                const int   ti = li[q];
                if (sw) { lc[q] = cc; li[q] = ci; cc = tf; ci = ti; }
            }
        }
    }

    __shared__ float s_iou[256 * KSEL];
    __shared__ float s_c[256 * KSEL];
    __shared__ int   s_i[256 * KSEL];
#pragma unroll
    for (int k = 0; k < KSEL; ++k) {
        s_iou[tid * KSEL + k] = liou[k];
        s_c[tid * KSEL + k]   = lc[k];
        s_i[tid * KSEL + k]   = li[k];
    }
    __syncthreads();

    for (int step = T >> 1; step >= 1; step >>= 1) {
        if (tid < step) {
            const int a = tid * KSEL;
            const int b = (tid + step) * KSEL;
            float mo[KSEL]; float mc[KSEL]; int mi[KSEL];
            int ia = 0, ib = 0;
#pragma unroll
            for (int k = 0; k < KSEL; ++k) {          // descending IoU merge
                const float va = s_iou[a + ia];
                const float vb = s_iou[b + ib];
                if (va >= vb) { mo[k] = va; ++ia; } else { mo[k] = vb; ++ib; }
            }
            ia = 0; ib = 0;
#pragma unroll
            for (int k = 0; k < KSEL; ++k) {          // ascending cost merge
                const float va = s_c[a + ia];
                const float vb = s_c[b + ib];
                if (va <= vb) { mc[k] = va; mi[k] = s_i[a + ia]; ++ia; }
                else          { mc[k] = vb; mi[k] = s_i[b + ib]; ++ib; }
            }
#pragma unroll
            for (int k = 0; k < KSEL; ++k) {
                s_iou[a + k] = mo[k];
                s_c[a + k]   = mc[k];
                s_i[a + k]   = mi[k];
            }
        }
        __syncthreads();
    }

    if (tid == 0) {
        float sum = 0.0f;
#pragma unroll
        for (int k = 0; k < KSEL; ++k) sum += s_iou[k];
        int dk = (int)sum;                 // truncation toward zero, as astype(int32)
        if (dk < 1) dk = 1;
        dyn_ks[gcol] = dk;
#pragma unroll
        for (int k = 0; k < KSEL; ++k) cand_idx[gcol * KSEL + k] = s_i[k];
    }
}

// ============================================================================
// Kernel 4: init per-prior match bookkeeping.
// ============================================================================
__global__ void init_kernel(int* __restrict__ match_count,
                            int* __restrict__ match_gt, int n) {
    const int i = blockIdx.x * blockDim.x + threadIdx.x;
    if (i < n) { match_count[i] = 0; match_gt[i] = 0x7fffffff; }
}

// ============================================================================
// Kernel 5: scatter candidate selections into per-prior counters.
// ============================================================================
__global__ void scatter_kernel(const int* __restrict__ dyn_ks,
                               const int* __restrict__ cand_idx,
                               int* __restrict__ match_count,
                               int* __restrict__ match_gt, int n) {
    const int gcol = blockIdx.x;
    const int k = threadIdx.x;
    if (k >= KSEL) return;
    int dk = dyn_ks[gcol];
    if (dk > KSEL) dk = KSEL;
    if (k < dk) {
        const int i = cand_idx[gcol * KSEL + k];
        if (i >= 0 && i < n) {
            atomicAdd(&match_count[i], 1);
            atomicMin(&match_gt[i], gcol);
        }
    }
}

// ============================================================================
// Kernel 6: finalize per prior (argmin-cost fallback for conflicts).
// ============================================================================
__global__ void finalize_kernel(const float* __restrict__ ious_t,
                                const float* __restrict__ cost_t,
                                const int* __restrict__ valid_mask,
                                const int* __restrict__ match_count,
                                const int* __restrict__ match_gt,
                                const int* __restrict__ labels,
                                int* __restrict__ out_inds,
                                float* __restrict__ out_iou,
                                int* __restrict__ out_lab,
                                int n, int g) {
    const int i = blockIdx.x * blockDim.x + threadIdx.x;
    if (i >= n) return;
    const int cnt = match_count[i];
    const bool fg = cnt > 0;
    int gi = 0;
    if (cnt == 1) {
        gi = match_gt[i];
    } else if (cnt > 1) {
        float best = 3e38f;
        for (int j = 0; j < g; ++j) {
            const float cv = cost_t[(size_t)j * (size_t)n + i];
            if (cv < best) { best = cv; gi = j; }
        }
    }
    float iou = 0.0f;
    if (fg) iou = ious_t[(size_t)gi * (size_t)n + i];
    const bool fin = fg && (valid_mask[i] != 0);
    out_inds[i] = fin ? gi + 1 : 0;
    out_iou[i]  = fin ? iou : -INF_COST;
    out_lab[i]  = fin ? labels[gi] : -1;
}

// ============================================================================
extern "C" void kernel_launch(void* const* d_in, const int* in_sizes, int n_in,
                              void* d_out, int out_size, void* d_ws, size_t ws_size,
                              hipStream_t stream) {
    const float* pred_scores = (const float*)d_in[0];   // n x c
    const float* priors      = (const float*)d_in[1];   // n x 4
    const float* pred_bboxes = (const float*)d_in[2];   // n x 4
    const float* gt_bboxes   = (const float*)d_in[3];   // g x 4
    const int*   gt_labels   = (const int*)d_in[4];     // g

    const int n = in_sizes[2] / 4;
    const int g = in_sizes[3] / 4;
    const int c = in_sizes[0] / n;
    const size_t nG = (size_t)n * (size_t)g;

    char* ws = (char*)d_ws;
    float* ious_t   = (float*)ws;                 ws += nG * sizeof(float);
    float* cost_t   = (float*)ws;                 ws += nG * sizeof(float);
    float* logits_t = (float*)ws;                 ws += nG * sizeof(float);
    int* valid_mask  = (int*)ws;                  ws += (size_t)n * sizeof(int);
    int* dyn_ks      = (int*)ws;                  ws += (size_t)g * sizeof(int);
    int* cand_idx    = (int*)ws;                  ws += (size_t)g * KSEL * sizeof(int);
    int* match_count = (int*)ws;                  ws += (size_t)n * sizeof(int);
    int* match_gt    = (int*)ws;

    int* out_inds  = (int*)d_out;
    float* out_iou = (float*)d_out + n;
    int* out_lab   = (int*)d_out + 2 * n;

    // 1) WMMA gather-GEMM for logits
    if (c == CDIM) {
        const int waves = n / 16;                       // one wave per 16-row tile
        const int blocks = (waves + 7) / 8;             // 8 waves / 256-thread block
        logits_wmma80_kernel<<<blocks, 256, 0, stream>>>(pred_scores, gt_labels,
                                                         logits_t, n, g);
    } else {
        const int tiles = (n / 16) * (g / 16);
        const int blocks = (tiles + 7) / 8;
        logits_wmma_kernel<<<blocks, 256, 0, stream>>>(pred_scores, gt_labels,
                                                       logits_t, n, g, c);
    }
    // 2) IoU / validity / cost (TDM-staged GT boxes)
    {
        const int blocks = (n + 255) / 256;
        cost_kernel<<<blocks, 256, 0, stream>>>(priors, pred_bboxes, gt_bboxes,
                                                logits_t, ious_t, cost_t,
                                                valid_mask, n, g);
    }
    // 3) per-GT top-k
    topk_kernel<<<g, 256, 0, stream>>>(ious_t, cost_t, dyn_ks, cand_idx, n);
    // 4) init match state
    init_kernel<<<(n + 255) / 256, 256, 0, stream>>>(match_count, match_gt, n);
    // 5) scatter selections
    scatter_kernel<<<g, 32, 0, stream>>>(dyn_ks, cand_idx, match_count, match_gt, n);
    // 6) finalize outputs
    finalize_kernel<<<(n + 255) / 256, 256, 0, stream>>>(
        ious_t, cost_t, valid_mask, match_count, match_gt, gt_labels,
        out_inds, out_iou, out_lab, n, g);
}